// GeoConv_20401094656096
// MI455X (gfx1250) — compile-verified
//
#include <hip/hip_runtime.h>
#include <math.h>

typedef __attribute__((ext_vector_type(2))) float v2f;
typedef __attribute__((ext_vector_type(8))) float v8f;

#define NTOT 16384   // B*n = 8*2048 nodes
#define CIN  64
#define HID  64
#define OUTC 128
#define KEDGE 16
#define LDA  392     // 6*64 edge-bucket features + 6 q scalars + 2 pad
#define BN_EPS 1e-5f

// ---------------------------------------------------------------------------
// Kernel 0: stack the 6 lins_W matrices + bias rows into Wcat (392 x 64).
// Rows 0..383 are lins_W flat, rows 384..389 are lins_b, rows 390..391 zero.
// ---------------------------------------------------------------------------
__global__ void build_wcat(const float* __restrict__ linsW,
                           const float* __restrict__ linsB,
                           float* __restrict__ Wc)
{
    int r = blockIdx.x;      // 0..391
    int h = threadIdx.x;     // 0..63
    float v = 0.0f;
    if (r < 384)       v = linsW[(size_t)r * 64 + h];
    else if (r < 390)  v = linsB[(size_t)(r - 384) * 64 + h];
    Wc[(size_t)r * 64 + h] = v;
}

// ---------------------------------------------------------------------------
// Kernel 1: per-node edge aggregation.
// One wave (32 lanes) per node; 8 nodes per 256-thread block.
// Produces A[t, s*64+c] = sum_e w[e]*cos2[d]*edge[e,c] for s=2d+(pdiff>0),
// and A[t, 384+s] = q[t,s] (bias coefficients), A[t,390..391]=0.
// ---------------------------------------------------------------------------
__global__ __launch_bounds__(256) void edge_aggregate(
    const float* __restrict__ x, const float* __restrict__ p,
    const int* __restrict__ sid, float* __restrict__ A)
{
    __shared__ float s_coef[8][16][3];
    __shared__ float s_pd[8][16];
    __shared__ int   s_sel[8][16];
    __shared__ int   s_sid[8][16];

    const int w    = threadIdx.x >> 5;
    const int lane = threadIdx.x & 31;
    const int t    = blockIdx.x * 8 + w;

    if (lane < 16) {
        int e  = t * KEDGE + lane;
        int sn = sid[e];
        s_sid[w][lane] = sn;
        float d0 = p[sn * 3 + 0] - p[t * 3 + 0];
        float d1 = p[sn * 3 + 1] - p[t * 3 + 1];
        float d2 = p[sn * 3 + 2] - p[t * 3 + 2];
        float dis = sqrtf(d0 * d0 + d1 * d1 + d2 * d2);
        dis = fmaxf(dis, 1e-16f);
        float inv = 1.0f / dis;
        float c0 = cosf(d0 * inv), c1 = cosf(d1 * inv), c2 = cosf(d2 * inv);
        s_coef[w][lane][0] = c0 * c0;
        s_coef[w][lane][1] = c1 * c1;
        s_coef[w][lane][2] = c2 * c2;
        s_sel[w][lane] = (d0 > 0.0f ? 1 : 0) | (d1 > 0.0f ? 2 : 0) | (d2 > 0.0f ? 4 : 0);
        s_pd[w][lane] = dis;
    }
    __syncthreads();

    // distance weights: w[e] = (1.1*max_k pd - pd[e])^2 / sum_k(...)
    float pr = 0.0f;
    for (int j = 0; j < 16; ++j) pr = fmaxf(pr, s_pd[w][j]);
    pr *= 1.1f;
    float sum = 0.0f;
    for (int j = 0; j < 16; ++j) { float v = pr - s_pd[w][j]; sum += v * v; }
    float invsum = 1.0f / sum;
    if (lane < 16) {
        float v   = pr - s_pd[w][lane];
        float wgt = v * v * invsum;
        s_coef[w][lane][0] *= wgt;
        s_coef[w][lane][1] *= wgt;
        s_coef[w][lane][2] *= wgt;
    }
    __syncthreads();

    // each lane owns 2 channels; 12 accumulators = 6 buckets x 2 channels
    const int c0  = lane * 2;
    float xt0 = x[(size_t)t * CIN + c0];
    float xt1 = x[(size_t)t * CIN + c0 + 1];
    float acc[12];
#pragma unroll
    for (int i = 0; i < 12; ++i) acc[i] = 0.0f;

    for (int e = 0; e < 16; ++e) {
        int sn = s_sid[w][e];
        float e0 = x[(size_t)sn * CIN + c0] - xt0;
        float e1 = x[(size_t)sn * CIN + c0 + 1] - xt1;
        int selb = s_sel[w][e];
#pragma unroll
        for (int d = 0; d < 3; ++d) {
            float cf  = s_coef[w][e][d];
            bool  hi  = (selb >> d) & 1;
            float clo = hi ? 0.0f : cf;
            float chi = hi ? cf : 0.0f;
            acc[(2 * d + 0) * 2 + 0] += clo * e0;
            acc[(2 * d + 0) * 2 + 1] += clo * e1;
            acc[(2 * d + 1) * 2 + 0] += chi * e0;
            acc[(2 * d + 1) * 2 + 1] += chi * e1;
        }
    }

    float* Arow = A + (size_t)t * LDA;
#pragma unroll
    for (int s = 0; s < 6; ++s) {
        Arow[s * 64 + c0]     = acc[s * 2 + 0];
        Arow[s * 64 + c0 + 1] = acc[s * 2 + 1];
    }
    if (lane < 6) {                 // bias coefficients q[t,s]
        int d = lane >> 1, m = lane & 1;
        float q = 0.0f;
        for (int e = 0; e < 16; ++e)
            if (((s_sel[w][e] >> d) & 1) == m) q += s_coef[w][e][d];
        Arow[384 + lane] = q;
    } else if (lane < 8) {
        Arow[384 + lane] = 0.0f;    // pad columns 390,391
    }
}

// ---------------------------------------------------------------------------
// Kernel 2: Y1[16384x64] = A[16384x392] @ Wcat[392x64] via V_WMMA_F32_16X16X4.
// One wave per 16x16 output tile; 4 waves/block cover the full N=64.
// A frag: lanes 0-15 hold K+0/K+1, lanes 16-31 hold K+2/K+3 (ISA 7.12.2).
// ---------------------------------------------------------------------------
__global__ __launch_bounds__(128) void gemm1_wmma(
    const float* __restrict__ A, const float* __restrict__ Wc,
    float* __restrict__ Y1)
{
    const int lane = threadIdx.x & 31;
    const int hf   = lane >> 4;
    const int l16  = lane & 15;
    const int n0   = (threadIdx.x >> 5) * 16;
    const int m0   = blockIdx.x * 16;

    v8f acc = {};
    const float* arow = A  + (size_t)(m0 + l16) * LDA + 2 * hf;
    const float* bcol = Wc + (size_t)(2 * hf) * HID + n0 + l16;
    for (int k = 0; k < LDA; k += 4) {
        v2f a = *(const v2f*)(arow + k);
        v2f b;
        b.x = bcol[(size_t)k * HID];
        b.y = bcol[(size_t)(k + 1) * HID];
        acc = __builtin_amdgcn_wmma_f32_16x16x4_f32(
            false, a, false, b, (short)0, acc, false, false);
    }
#pragma unroll
    for (int r = 0; r < 8; ++r)
        Y1[(size_t)(m0 + r + 8 * hf) * HID + n0 + l16] = acc[r];
}

// ---------------------------------------------------------------------------
// Kernel 3: per-column sum / sumsq reduction (for batchnorm stats).
// ---------------------------------------------------------------------------
template <int C>
__global__ __launch_bounds__(256) void colreduce(
    const float* __restrict__ X, int rowsPerBlock,
    float* __restrict__ sums, float* __restrict__ sqs)
{
    __shared__ float ls[256], lq[256];
    const int ch   = threadIdx.x % C;
    const int r0   = threadIdx.x / C;
    const int base = blockIdx.x * rowsPerBlock;
    float s = 0.0f, q = 0.0f;
    for (int r = r0; r < rowsPerBlock; r += 256 / C) {
        float v = X[(size_t)(base + r) * C + ch];
        s += v; q += v * v;
    }
    ls[threadIdx.x] = s; lq[threadIdx.x] = q;
    __syncthreads();
    if (threadIdx.x < C) {
        for (int i = threadIdx.x + C; i < 256; i += C) { s += ls[i]; q += lq[i]; }
        atomicAdd(&sums[ch], s);
        atomicAdd(&sqs[ch], q);
    }
}

__global__ void finalize_bn(const float* __restrict__ sums,
                            const float* __restrict__ sqs,
                            const float* __restrict__ g, const float* __restrict__ b,
                            int C, float invN,
                            float* __restrict__ scale, float* __restrict__ shift)
{
    int c = blockIdx.x * blockDim.x + threadIdx.x;
    if (c < C) {
        float mean = sums[c] * invN;
        float var  = sqs[c] * invN - mean * mean;
        float rs   = rsqrtf(var + BN_EPS);
        float sc   = g[c] * rs;
        scale[c] = sc;
        shift[c] = b[c] - mean * sc;
    }
}

// ---------------------------------------------------------------------------
// Kernel 4: Z[16384x128] = x@lin1_W + relu(bn1(Y1))@lin2_W + (lin1_b+lin2_b).
// BN1+ReLU fused into the A-operand load of the second K loop.
// ---------------------------------------------------------------------------
__global__ __launch_bounds__(256) void gemm2_wmma(
    const float* __restrict__ x, const float* __restrict__ Y1,
    const float* __restrict__ W1, const float* __restrict__ W2,
    const float* __restrict__ b1v, const float* __restrict__ b2v,
    const float* __restrict__ scale1, const float* __restrict__ shift1,
    float* __restrict__ Z)
{
    const int lane = threadIdx.x & 31;
    const int hf   = lane >> 4;
    const int l16  = lane & 15;
    const int n0   = (threadIdx.x >> 5) * 16;   // 8 waves -> N=128
    const int m0   = blockIdx.x * 16;
    const int row  = m0 + l16;

    v8f acc = {};
    for (int k = 0; k < CIN; k += 4) {
        int ka = k + 2 * hf;
        v2f a = *(const v2f*)(x + (size_t)row * CIN + ka);
        v2f b;
        b.x = W1[(size_t)ka * OUTC + n0 + l16];
        b.y = W1[(size_t)(ka + 1) * OUTC + n0 + l16];
        acc = __builtin_amdgcn_wmma_f32_16x16x4_f32(
            false, a, false, b, (short)0, acc, false, false);
    }
    for (int k = 0; k < HID; k += 4) {
        int ka = k + 2 * hf;
        float y0 = Y1[(size_t)row * HID + ka];
        float y1 = Y1[(size_t)row * HID + ka + 1];
        v2f a;
        a.x = fmaxf(y0 * scale1[ka]     + shift1[ka],     0.0f);
        a.y = fmaxf(y1 * scale1[ka + 1] + shift1[ka + 1], 0.0f);
        v2f b;
        b.x = W2[(size_t)ka * OUTC + n0 + l16];
        b.y = W2[(size_t)(ka + 1) * OUTC + n0 + l16];
        acc = __builtin_amdgcn_wmma_f32_16x16x4_f32(
            false, a, false, b, (short)0, acc, false, false);
    }
    float bias = b1v[n0 + l16] + b2v[n0 + l16];
#pragma unroll
    for (int r = 0; r < 8; ++r)
        Z[(size_t)(m0 + r + 8 * hf) * OUTC + n0 + l16] = acc[r] + bias;
}

// Kernel 5: out = relu(bn2(Z)) in place on d_out.
__global__ __launch_bounds__(256) void bn_relu_out(
    float* __restrict__ Z,
    const float* __restrict__ scale, const float* __restrict__ shift)
{
    int idx = blockIdx.x * 256 + threadIdx.x;
    int c   = idx & (OUTC - 1);
    float v = Z[idx];
    Z[idx] = fmaxf(v * scale[c] + shift[c], 0.0f);
}

// ---------------------------------------------------------------------------
extern "C" void kernel_launch(void* const* d_in, const int* in_sizes, int n_in,
                              void* d_out, int out_size, void* d_ws, size_t ws_size,
                              hipStream_t stream)
{
    const float* x     = (const float*)d_in[0];
    const float* p     = (const float*)d_in[1];
    const int*   sid   = (const int*)d_in[2];
    // d_in[3] tid_euc (implicit arange-repeat), d_in[4]=B, d_in[5]=n : constants
    const float* linsW = (const float*)d_in[6];
    const float* linsB = (const float*)d_in[7];
    const float* lin1W = (const float*)d_in[8];
    const float* lin1b = (const float*)d_in[9];
    const float* lin2W = (const float*)d_in[10];
    const float* lin2b = (const float*)d_in[11];
    const float* g1    = (const float*)d_in[12];
    const float* b1    = (const float*)d_in[13];
    const float* g2    = (const float*)d_in[14];
    const float* b2    = (const float*)d_in[15];

    // workspace layout (floats); total ~28.6 MB
    float* ws    = (float*)d_ws;
    float* A     = ws;                                   // NTOT*LDA
    float* Wc    = A + (size_t)NTOT * LDA;               // 392*64
    float* Y1    = Wc + 392 * 64;                        // NTOT*HID
    float* stats = Y1 + (size_t)NTOT * HID;              // 768 floats
    float* sum1   = stats;         // 64
    float* sq1    = stats + 64;    // 64
    float* sum2   = stats + 128;   // 128
    float* sq2    = stats + 256;   // 128
    float* scale1 = stats + 384;   // 64
    float* shift1 = stats + 448;   // 64
    float* scale2 = stats + 512;   // 128
    float* shift2 = stats + 640;   // 128

    float* Z = (float*)d_out;

    // zero the atomic accumulators every call (graph-capture safe)
    hipMemsetAsync(stats, 0, 384 * sizeof(float), stream);

    build_wcat<<<392, 64, 0, stream>>>(linsW, linsB, Wc);
    edge_aggregate<<<NTOT / 8, 256, 0, stream>>>(x, p, sid, A);
    gemm1_wmma<<<NTOT / 16, 128, 0, stream>>>(A, Wc, Y1);
    colreduce<64><<<64, 256, 0, stream>>>(Y1, NTOT / 64, sum1, sq1);
    finalize_bn<<<1, 64, 0, stream>>>(sum1, sq1, g1, b1, 64, 1.0f / NTOT, scale1, shift1);
    gemm2_wmma<<<NTOT / 16, 256, 0, stream>>>(x, Y1, lin1W, lin2W, lin1b, lin2b,
                                              scale1, shift1, Z);
    colreduce<128><<<64, 256, 0, stream>>>(Z, NTOT / 64, sum2, sq2);
    finalize_bn<<<1, 128, 0, stream>>>(sum2, sq2, g2, b2, 128, 1.0f / NTOT, scale2, shift2);
    bn_relu_out<<<(NTOT * OUTC) / 256, 256, 0, stream>>>(Z, scale2, shift2);
}